// LSTMSparseReservoir_1245540516183
// MI455X (gfx1250) — compile-verified
//
#include <hip/hip_runtime.h>
#include <hip/hip_bf16.h>
#include <cstdint>

// Problem constants (from reference)
#define NN     16384    // N (columns/rows of sparse operator)
#define BB     512      // batch
#define NNZ    131072
#define GRP    16       // columns per workgroup (WMMA M dimension)
#define NGROUP (NN / GRP)          // 1024 groups
#define GCAP   256      // per-group edge capacity (Poisson lambda=128; P(>256) ~ 0)
#define CHUNK  16       // TDM gather-mode: 16 row indices (16-bit) per descriptor

typedef unsigned int u32x4 __attribute__((ext_vector_type(4)));
typedef int          i32x8 __attribute__((ext_vector_type(8)));
typedef int          i32x4 __attribute__((ext_vector_type(4)));
typedef float        v2f   __attribute__((ext_vector_type(2)));
typedef float        v8f   __attribute__((ext_vector_type(8)));

// ---------------------------------------------------------------------------
// TDM gather-mode descriptor: gather 16 rows (each BB contiguous f32) of the
// (NN x BB) tensor at gbase into LDS at lds_addr. Indices are 16 u16 values
// at idx16 (16B-aligned). CDNA5 ISA. 6-arg builtin form.
// ---------------------------------------------------------------------------
__device__ __forceinline__ void tdm_gather16(unsigned lds_addr,
                                             const float* __restrict__ gbase,
                                             const unsigned short* __restrict__ idx16)
{
    unsigned long long ga = (unsigned long long)(size_t)(const void*)gbase;

    u32x4 g0;
    g0[0] = 1u | (1u << 31);                 // count=1, gather_mode=1, 16-bit idx
    g0[1] = lds_addr;                        // LDS byte address
    g0[2] = (unsigned)(ga & 0xFFFFFFFFull);  // global_addr[31:0]
    g0[3] = (unsigned)((ga >> 32) & 0x1FFFFFFull) | (2u << 30); // addr[56:32] | type=2

    i32x8 g1;
    g1[0] = (int)(2u << 16);                 // wg_mask=0, data_size=2 (4B)
    g1[1] = (int)(512u << 16);               // tensor_dim0[15:0]=512 (row length BB)
    g1[2] = (int)(16384u << 16);             // dim0[31:16]=0 | tensor_dim1[15:0]=NN
    g1[3] = (int)(512u << 16);               // dim1[31:16]=0 | tile_dim0=512
    g1[4] = 16;                              // tile_dim1 = #valid indices = 16
    g1[5] = 512;                             // tensor_dim0_stride = 512
    g1[6] = 0;
    g1[7] = 0;

    const i32x4* ip = reinterpret_cast<const i32x4*>(idx16);
    i32x4 g2 = ip[0];                        // row_index_0..7  (2 u16 per dword)
    i32x4 g3 = ip[1];                        // row_index_8..15

    i32x8 gx;                                // extra operand in 6-arg form: zeros
    gx[0] = 0; gx[1] = 0; gx[2] = 0; gx[3] = 0;
    gx[4] = 0; gx[5] = 0; gx[6] = 0; gx[7] = 0;

    __builtin_amdgcn_tensor_load_to_lds(g0, g1, g2, g3, gx, 0);
}

__device__ __forceinline__ float sigmoid_f(float x) {
    return __builtin_amdgcn_rcpf(1.0f + __expf(-x));
}
__device__ __forceinline__ float tanh_f(float x) {
    float ax = fabsf(x);
    float e  = __expf(-2.0f * ax);                       // in (0,1], no overflow
    float t  = (1.0f - e) * __builtin_amdgcn_rcpf(1.0f + e);
    return copysignf(t, x);
}

// ---------------------------------------------------------------------------
// Transpose: in is (R x C) row-major -> out is (C x R). grid(C/32,R/32), block(32,32).
// ---------------------------------------------------------------------------
__global__ void transpose_k(const float* __restrict__ in, float* __restrict__ out,
                            int R, int C)
{
    __shared__ float t[32][33];
    int c0 = blockIdx.x * 32, r0 = blockIdx.y * 32;
    t[threadIdx.y][threadIdx.x] = in[(size_t)(r0 + threadIdx.y) * C + c0 + threadIdx.x];
    __syncthreads();
    out[(size_t)(c0 + threadIdx.y) * R + r0 + threadIdx.x] = t[threadIdx.x][threadIdx.y];
}

// ---------------------------------------------------------------------------
// Build per-GROUP edge slots (group = 16 columns): row index (u16, feeds TDM
// gather), value (f32, zero-padded), owner column-in-group (u8).
// ---------------------------------------------------------------------------
__global__ void fill_edges(const long long* __restrict__ sp,
                           const float* __restrict__ kern,
                           int* __restrict__ cntg,
                           unsigned short* __restrict__ grow,
                           float* __restrict__ gval,
                           unsigned char* __restrict__ gown)
{
    int k = blockIdx.x * blockDim.x + threadIdx.x;
    if (k >= NNZ) return;
    int r = (int)sp[2 * k + 0];
    int c = (int)sp[2 * k + 1];
    int g = c >> 4;
    int p = atomicAdd(&cntg[g], 1);
    if (p < GCAP) {
        grow[(size_t)g * GCAP + p] = (unsigned short)r;
        gval[(size_t)g * GCAP + p] = kern[k];
        gown[(size_t)g * GCAP + p] = (unsigned char)(c & 15);
    }
}

// ---------------------------------------------------------------------------
// One fused reservoir-LSTM iteration. One workgroup (256 thr = 8 waves) per
// group of 16 columns:
//   TDM gather-mode pulls 16 h rows/chunk (double-buffered LDS, TENSORcnt
//   pipelined); Z(16x512) = A(16xK) x B(Kx512) accumulated on the matrix pipe
//   with V_WMMA_F32_16X16X4_F32 (A[c,k] = val_k if owner(k)==c else 0);
//   LSTM gates run on the accumulator layout (VALU co-executes with XDL).
// ---------------------------------------------------------------------------
__global__ __launch_bounds__(256) void lstm_iter(
    const float* __restrict__ h_in,   // (NN x BB)
    float* __restrict__ h_out,        // (NN x BB)
    float* __restrict__ c_st,         // (NN x BB), in-place
    const unsigned short* __restrict__ grow,
    const float* __restrict__ gval,
    const unsigned char* __restrict__ gown,
    const int* __restrict__ cntg,
    const float* __restrict__ bi, const float* __restrict__ bf,
    const float* __restrict__ bo, const float* __restrict__ bg)
{
    __shared__ float tile[2][CHUNK * BB];   // 2 x 32 KB double buffer
    __shared__ float sbias[64];             // 4 gates x 16 columns

    const int g     = blockIdx.x;
    const int cbase = g * GRP;
    const int tid   = threadIdx.x;
    const int lane  = tid & 31;
    const int half  = lane >> 4;            // 0 | 1 (wave32 half)
    const int m     = lane & 15;            // M (A row / C-D column N index)
    const int wv    = tid >> 5;             // wave id 0..7

    int n = cntg[g];
    if (n > GCAP) n = GCAP;
    const int nch = (n + CHUNK - 1) >> 4;   // 0..16 chunks (padded slots: val=0)

    if (tid < 64) {                         // stage gate biases for this group
        const float* src = (tid < 16) ? bi : (tid < 32) ? bf : (tid < 48) ? bo : bg;
        sbias[tid] = src[cbase + (tid & 15)];
    }

    const unsigned lds0      = (unsigned)(size_t)(void*)&tile[0][0];
    const unsigned ldsStride = (unsigned)(CHUNK * BB * sizeof(float));
    const size_t   ebase     = (size_t)g * GCAP;

    if (nch > 0 && tid < 32)                // wave 0 drives the TDM
        tdm_gather16(lds0, h_in, grow + ebase);
    __syncthreads();                        // sbias visible even when nch==0

    v8f acc[4];                             // 4 N-tiles of Z (16 cols x 16 b)
    #pragma unroll
    for (int t = 0; t < 4; ++t) acc[t] = v8f{0.f,0.f,0.f,0.f,0.f,0.f,0.f,0.f};

    for (int k = 0; k < nch; ++k) {
        if (tid < 32) {
            if (k + 1 < nch) {
                tdm_gather16(lds0 + (unsigned)((k + 1) & 1) * ldsStride,
                             h_in, grow + ebase + (size_t)(k + 1) * CHUNK);
                __builtin_amdgcn_s_wait_tensorcnt(1);   // chunk k landed
            } else {
                __builtin_amdgcn_s_wait_tensorcnt(0);   // all landed
            }
        }
        __syncthreads();                                // publish LDS tile

        const size_t eb = ebase + (size_t)k * CHUNK;
        const float* __restrict__ tl = &tile[k & 1][0];

        // A fragments for this 16-edge chunk (shared across the 4 N-tiles).
        // 16x4 f32 A layout: lane m holds M=m; VGPR0 = K 0/2 (by half), VGPR1 = K 1/3.
        v2f afr[4];
        #pragma unroll
        for (int j = 0; j < 4; ++j) {
            int e0 = 4 * j + 2 * half;
            unsigned char o0 = gown[eb + e0], o1 = gown[eb + e0 + 1];
            float v0 = gval[eb + e0], v1 = gval[eb + e0 + 1];
            afr[j].x = (o0 == (unsigned char)m) ? v0 : 0.0f;
            afr[j].y = (o1 == (unsigned char)m) ? v1 : 0.0f;
        }

        #pragma unroll
        for (int t = 0; t < 4; ++t) {
            const int nb = wv * 64 + t * 16;            // batch base of this N-tile
            #pragma unroll
            for (int j = 0; j < 4; ++j) {
                // B 4x16 f32: VGPR0 = rows K 0/2 (by half), VGPR1 = rows K 1/3; N = lane%16
                int r0 = 4 * j + 2 * half;
                v2f bfr;
                bfr.x = tl[(r0    ) * BB + nb + m];
                bfr.y = tl[(r0 + 1) * BB + nb + m];
                acc[t] = __builtin_amdgcn_wmma_f32_16x16x4_f32(
                             false, afr[j], false, bfr,
                             (short)0, acc[t], false, false);
            }
        }
        __syncthreads();                                // tile[k&1] reusable at k+2
    }

    // Gates on the C/D layout: VGPR r <-> column M = r + 8*half, N(batch) = m.
    #pragma unroll
    for (int t = 0; t < 4; ++t) {
        const int nb = wv * 64 + t * 16;
        #pragma unroll
        for (int r = 0; r < 8; ++r) {
            const int   mrow = r + 8 * half;
            const float z    = acc[t][r];
            const size_t off = (size_t)(cbase + mrow) * BB + (size_t)(nb + m);
            float i = sigmoid_f(z + sbias[ 0 + mrow]);
            float f = sigmoid_f(z + sbias[16 + mrow]);
            float o = sigmoid_f(z + sbias[32 + mrow]);
            float gg = tanh_f  (z + sbias[48 + mrow]);
            float cn = f * c_st[off] + i * gg;
            c_st[off]  = cn;
            h_out[off] = o * tanh_f(cn);
        }
    }
}

// ---------------------------------------------------------------------------
// Orchestration. Workspace (~102.5 MB, fits the 192 MB L2):
//   hA, hB : 2 x NN*BB f32 ping-pong h buffers    (67 MB)
//   cst    : NN*BB f32 cell state                 (33.5 MB)
//   gval   : NGROUP*GCAP f32 edge values (0-pad)  (1 MB)
//   grow   : NGROUP*GCAP u16 edge row indices     (512 KB)
//   gown   : NGROUP*GCAP u8 owner col-in-group    (256 KB)
//   cntg   : NGROUP i32 per-group edge counts     (4 KB)
// ---------------------------------------------------------------------------
extern "C" void kernel_launch(void* const* d_in, const int* in_sizes, int n_in,
                              void* d_out, int out_size, void* d_ws, size_t ws_size,
                              hipStream_t stream)
{
    const float*     x    = (const float*)d_in[0];
    const float*     kern = (const float*)d_in[1];
    const float*     b_i  = (const float*)d_in[2];
    const float*     b_f  = (const float*)d_in[3];
    const float*     b_o  = (const float*)d_in[4];
    const float*     b_g  = (const float*)d_in[5];
    const long long* sp   = (const long long*)d_in[6];

    char* w = (char*)d_ws;
    float* hA = (float*)w;                      w += (size_t)NN * BB * sizeof(float);
    float* hB = (float*)w;                      w += (size_t)NN * BB * sizeof(float);
    float* cst = (float*)w;                     w += (size_t)NN * BB * sizeof(float);
    float* gval = (float*)w;                    w += (size_t)NGROUP * GCAP * sizeof(float);
    unsigned short* grow = (unsigned short*)w;  w += (size_t)NGROUP * GCAP * sizeof(unsigned short);
    unsigned char* gown = (unsigned char*)w;    w += (size_t)NGROUP * GCAP * sizeof(unsigned char);
    int* cntg = (int*)w;                        w += (size_t)NGROUP * sizeof(int);

    (void)hipMemsetAsync(cst,  0, (size_t)NN * BB * sizeof(float),                stream);
    (void)hipMemsetAsync(gval, 0, (size_t)NGROUP * GCAP * sizeof(float),          stream);
    (void)hipMemsetAsync(grow, 0, (size_t)NGROUP * GCAP * sizeof(unsigned short), stream);
    (void)hipMemsetAsync(gown, 0, (size_t)NGROUP * GCAP * sizeof(unsigned char),  stream);
    (void)hipMemsetAsync(cntg, 0, (size_t)NGROUP * sizeof(int),                   stream);

    dim3 tb(32, 32);
    // x (BB x NN) -> hA (NN x BB): a logical column becomes a contiguous 2KB row
    transpose_k<<<dim3(NN / 32, BB / 32), tb, 0, stream>>>(x, hA, BB, NN);
    fill_edges<<<NNZ / 256, 256, 0, stream>>>(sp, kern, cntg, grow, gval, gown);

    float* hin = hA;
    float* hout = hB;
    for (int r = 0; r < 4; ++r) {               // R+1 = 4 iterations
        lstm_iter<<<NGROUP, 256, 0, stream>>>(hin, hout, cst, grow, gval, gown,
                                              cntg, b_i, b_f, b_o, b_g);
        float* t = hin; hin = hout; hout = t;
    }
    // After 4 swaps the final h is in hin (== hA). Back to (BB x NN).
    transpose_k<<<dim3(BB / 32, NN / 32), tb, 0, stream>>>(hin, (float*)d_out, NN, BB);
}